// LearnableKalmanFilter_19567871001295
// MI455X (gfx1250) — compile-verified
//
#include <hip/hip_runtime.h>
#include <math.h>

// CDNA5 fp32 WMMA: D(16x16) = A(16x4) * B(4x16) + C, wave32.
typedef float v2f __attribute__((ext_vector_type(2)));
typedef float v8f __attribute__((ext_vector_type(8)));

#define KF_EPS 1e-6f

// ---- WMMA operand builders (layouts per CDNA5 ISA 7.12.2) ----
// A 16x4 f32: VGPR0: K=0 (lanes 0-15) / K=2 (lanes 16-31); VGPR1: K=1 / K=3.
static __device__ __forceinline__ v2f lds_a_chunk(const float* buf, int kc, int lane) {
  const int m  = lane & 15;
  const int kb = kc * 4 + ((lane >> 4) << 1);
  v2f a;
  a.x = buf[m * 16 + kb];
  a.y = buf[m * 16 + kb + 1];
  return a;
}

// B 4x16 f32: VGPR0: K=0 (lanes 0-15) / K=2 (lanes 16-31); VGPR1: K=1 / K=3.
template <bool TRANS>
static __device__ __forceinline__ v2f lds_b_chunk(const float* buf, int kc, int lane) {
  const int n  = lane & 15;
  const int kb = kc * 4 + ((lane >> 4) << 1);
  v2f b;
  if (TRANS) {            // B = buf^T : B(k,n) = buf[n][k]
    b.x = buf[n * 16 + kb];
    b.y = buf[n * 16 + kb + 1];
  } else {                // B(k,n) = buf[k][n]
    b.x = buf[kb * 16 + n];
    b.y = buf[(kb + 1) * 16 + n];
  }
  return b;
}

// Full 16x16x16 product as 4 chained K=4 WMMAs.
template <bool TRANS>
static __device__ __forceinline__ v8f mm16(const float* Abuf, const float* Bbuf, int lane) {
  v8f acc = {0.f, 0.f, 0.f, 0.f, 0.f, 0.f, 0.f, 0.f};
#pragma unroll
  for (int kc = 0; kc < 4; ++kc) {
    v2f a = lds_a_chunk(Abuf, kc, lane);
    v2f b = lds_b_chunk<TRANS>(Bbuf, kc, lane);
    acc = __builtin_amdgcn_wmma_f32_16x16x4_f32(false, a, false, b, (short)0, acc,
                                                false, false);
  }
  return acc;
}

// C/D 16x16 f32 layout: VGPR i: row i (lanes 0-15) / row i+8 (lanes 16-31), col = lane%16.
static __device__ __forceinline__ void store_d(float* buf, v8f d, int lane) {
  const int n  = lane & 15;
  const int mb = (lane >> 4) * 8;
#pragma unroll
  for (int i = 0; i < 8; ++i) buf[(mb + i) * 16 + n] = d[i];
}

// Standard adjugate 4x4 inverse (S is symmetric PD so orientation is irrelevant).
static __device__ __forceinline__ void inv4x4(const float m[16], float inv[16]) {
  inv[0]  =  m[5]*m[10]*m[15] - m[5]*m[11]*m[14] - m[9]*m[6]*m[15] + m[9]*m[7]*m[14] + m[13]*m[6]*m[11] - m[13]*m[7]*m[10];
  inv[4]  = -m[4]*m[10]*m[15] + m[4]*m[11]*m[14] + m[8]*m[6]*m[15] - m[8]*m[7]*m[14] - m[12]*m[6]*m[11] + m[12]*m[7]*m[10];
  inv[8]  =  m[4]*m[9]*m[15]  - m[4]*m[11]*m[13] - m[8]*m[5]*m[15] + m[8]*m[7]*m[13] + m[12]*m[5]*m[11] - m[12]*m[7]*m[9];
  inv[12] = -m[4]*m[9]*m[14]  + m[4]*m[10]*m[13] + m[8]*m[5]*m[14] - m[8]*m[6]*m[13] - m[12]*m[5]*m[10] + m[12]*m[6]*m[9];
  inv[1]  = -m[1]*m[10]*m[15] + m[1]*m[11]*m[14] + m[9]*m[2]*m[15] - m[9]*m[3]*m[14] - m[13]*m[2]*m[11] + m[13]*m[3]*m[10];
  inv[5]  =  m[0]*m[10]*m[15] - m[0]*m[11]*m[14] - m[8]*m[2]*m[15] + m[8]*m[3]*m[14] + m[12]*m[2]*m[11] - m[12]*m[3]*m[10];
  inv[9]  = -m[0]*m[9]*m[15]  + m[0]*m[11]*m[13] + m[8]*m[1]*m[15] - m[8]*m[3]*m[13] - m[12]*m[1]*m[11] + m[12]*m[3]*m[9];
  inv[13] =  m[0]*m[9]*m[14]  - m[0]*m[10]*m[13] - m[8]*m[1]*m[14] + m[8]*m[2]*m[13] + m[12]*m[1]*m[10] - m[12]*m[2]*m[9];
  inv[2]  =  m[1]*m[6]*m[15]  - m[1]*m[7]*m[14]  - m[5]*m[2]*m[15] + m[5]*m[3]*m[14] + m[13]*m[2]*m[7]  - m[13]*m[3]*m[6];
  inv[6]  = -m[0]*m[6]*m[15]  + m[0]*m[7]*m[14]  + m[4]*m[2]*m[15] - m[4]*m[3]*m[14] - m[12]*m[2]*m[7]  + m[12]*m[3]*m[6];
  inv[10] =  m[0]*m[5]*m[15]  - m[0]*m[7]*m[13]  - m[4]*m[1]*m[15] + m[4]*m[3]*m[13] + m[12]*m[1]*m[7]  - m[12]*m[3]*m[5];
  inv[14] = -m[0]*m[5]*m[14]  + m[0]*m[6]*m[13]  + m[4]*m[1]*m[14] - m[4]*m[2]*m[13] - m[12]*m[1]*m[6]  + m[12]*m[2]*m[5];
  inv[3]  = -m[1]*m[6]*m[11]  + m[1]*m[7]*m[10]  + m[5]*m[2]*m[11] - m[5]*m[3]*m[10] - m[9]*m[2]*m[7]   + m[9]*m[3]*m[6];
  inv[7]  =  m[0]*m[6]*m[11]  - m[0]*m[7]*m[10]  - m[4]*m[2]*m[11] + m[4]*m[3]*m[10] + m[8]*m[2]*m[7]   - m[8]*m[3]*m[6];
  inv[11] = -m[0]*m[5]*m[11]  + m[0]*m[7]*m[9]   + m[4]*m[1]*m[11] - m[4]*m[3]*m[9]  - m[8]*m[1]*m[7]   + m[8]*m[3]*m[5];
  inv[15] =  m[0]*m[5]*m[10]  - m[0]*m[6]*m[9]   - m[4]*m[1]*m[10] + m[4]*m[2]*m[9]  + m[8]*m[1]*m[6]   - m[8]*m[2]*m[5];
  const float det = m[0]*inv[0] + m[1]*inv[4] + m[2]*inv[8] + m[3]*inv[12];
  const float rd  = 1.0f / det;
#pragma unroll
  for (int i = 0; i < 16; ++i) inv[i] *= rd;
}

// One wave = 2 tracks packed block-diagonally into 16x16 WMMA tiles.
// Block = 256 threads = 8 waves = 16 tracks.
__global__ __launch_bounds__(256) void kalman_wmma_kernel(
    const float* __restrict__ g_state,  // [B,8]
    const float* __restrict__ g_cov,    // [B,64]
    const float* __restrict__ g_meas,   // [B,4]
    const float* __restrict__ g_F,      // [8,8]
    const float* __restrict__ g_H,      // [4,8]
    const float* __restrict__ g_lq,     // [8]
    const float* __restrict__ g_lr,     // [4]
    float* __restrict__ o_state,        // [B,8]
    float* __restrict__ o_cov,          // [B,64]
    int B) {
  __shared__ float sF[256];        // diag(F,F), row-major 16x16
  __shared__ float sH[256];        // diag(H,H): rows 0-3 / 8-11
  __shared__ float sqq[8];
  __shared__ float srr[4];
  __shared__ float bP[8][256];     // per-wave: P_big, later A_big = I-KH
  __shared__ float bT[8][256];     // per-wave: T=F*P -> U=H*Sigma -> V=A*Sigma -> cov staging
  __shared__ float bC[8][256];     // per-wave: Sigma (cov_pred)
  __shared__ float sS[8][32];
  __shared__ float sK[8][64];
  __shared__ float ssp[8][16];
  __shared__ float syv[8][8];
  __shared__ float sst[8][16];
  __shared__ float smz[8][8];

  const int tid  = threadIdx.x;
  const int lane = tid & 31;
  const int wv   = tid >> 5;
  const int hi   = lane >> 4;      // 0: track A half, 1: track B half
  const int ln   = lane & 15;

  // ---- shared init: block-diagonal F/H tiles, exp(log_q)+eps, exp(log_r)+eps ----
  {
    const int m = tid >> 4, k = tid & 15;
    float fv = 0.f, hv = 0.f;
    if (m < 8 && k < 8)               fv = g_F[m * 8 + k];
    else if (m >= 8 && k >= 8)        fv = g_F[(m - 8) * 8 + (k - 8)];
    if (m < 4 && k < 8)               hv = g_H[m * 8 + k];
    else if (m >= 8 && m < 12 && k >= 8) hv = g_H[(m - 8) * 8 + (k - 8)];
    sF[tid] = fv;
    sH[tid] = hv;
    if (tid < 8) sqq[tid] = expf(g_lq[tid]) + KF_EPS;
    if (tid < 4) srr[tid] = expf(g_lr[tid]) + KF_EPS;
  }

  int t0 = ((int)blockIdx.x * 8 + wv) * 2;         // first of 2 consecutive tracks
  if (t0 + 1 >= B) t0 = (B >= 2) ? (B - 2) : 0;    // safety clamp (B=262144 is exact)

  float* P = bP[wv];
  float* T = bT[wv];
  float* C = bC[wv];

  // zero P_big, then fill the two diagonal 8x8 blocks with a coalesced float4 load
#pragma unroll
  for (int j = 0; j < 8; ++j) P[lane * 8 + j] = 0.f;
  {
    const float4 v = reinterpret_cast<const float4*>(g_cov + (size_t)t0 * 64)[lane];
    const int g = lane * 4, t = g >> 6, m = (g >> 3) & 7, n0 = g & 7;
    float* dst = &P[(m + 8 * t) * 16 + 8 * t + n0];
    dst[0] = v.x; dst[1] = v.y; dst[2] = v.z; dst[3] = v.w;
  }
  if (lane < 16) sst[wv][lane] = g_state[(size_t)t0 * 8 + lane];
  if (lane < 8)  smz[wv][lane] = g_meas[(size_t)t0 * 4 + lane];
  __syncthreads();

  v8f acc;

  // ---- T = F_big * P_big ----
  acc = mm16<false>(sF, P, lane);
  store_d(T, acc, lane);
  __syncthreads();

  // ---- Sigma = F_big * T^T + Q   (valid since P symmetric: F(FP)^T = F P F^T) ----
  acc = mm16<true>(sF, T, lane);
#pragma unroll
  for (int i = 0; i < 8; ++i)
    if (ln == i + 8 * hi) acc[i] += sqq[i];        // diagonal m==n, q index = m%8
  store_d(C, acc, lane);
  __syncthreads();

  // ---- U = H_big * Sigma ----
  acc = mm16<false>(sH, C, lane);
  store_d(T, acc, lane);
  __syncthreads();

  // ---- S = U * H^T + R : one element per lane (track = hi, i = ln/4, j = ln%4) ----
  {
    const int t = hi, i = ln >> 2, j = ln & 3;
    float s = (i == j) ? srr[i] : 0.f;
#pragma unroll
    for (int a = 0; a < 8; ++a)
      s += T[(i + 8 * t) * 16 + (a + 8 * t)] * sH[(j + 8 * t) * 16 + (a + 8 * t)];
    sS[wv][t * 16 + (i * 4 + j)] = s;
  }
  __syncthreads();

  // ---- S^-1 (each lane computes its track's inverse redundantly; ~100 flops) ----
  float si[16];
  {
    float s[16];
#pragma unroll
    for (int e = 0; e < 16; ++e) s[e] = sS[wv][hi * 16 + e];
    inv4x4(s, si);
  }

  // ---- K = U^T * S^-1  (Sigma symmetric => U^T = Sigma H^T); 2 elems/lane ----
  {
    const int t = hi;
#pragma unroll
    for (int e = 0; e < 2; ++e) {
      const int idx = ln * 2 + e, m = idx >> 2, j = idx & 3;
      float kv = 0.f;
#pragma unroll
      for (int a = 0; a < 4; ++a)
        kv += T[(a + 8 * t) * 16 + (m + 8 * t)] * si[a * 4 + j];
      sK[wv][t * 32 + idx] = kv;
    }
  }

  // ---- state predict: sp = F s ----
  if (ln < 8) {
    const int t = hi, m = ln;
    float sp = 0.f;
#pragma unroll
    for (int j = 0; j < 8; ++j) sp += sF[m * 16 + j] * sst[wv][t * 8 + j];
    ssp[wv][t * 8 + m] = sp;
  }
  __syncthreads();

  // ---- innovation: y = z - H sp ----
  if (ln < 4) {
    const int t = hi, m = ln;
    float yv = smz[wv][t * 4 + m];
#pragma unroll
    for (int a = 0; a < 8; ++a)
      yv -= sH[(m + 8 * t) * 16 + (a + 8 * t)] * ssp[wv][t * 8 + a];
    syv[wv][t * 4 + m] = yv;
  }
  __syncthreads();

  // ---- state update (global write: lanes 0-7 -> t0, 16-23 -> t0+1, contiguous) ----
  if (ln < 8) {
    const int t = hi, m = ln;
    float su = ssp[wv][t * 8 + m];
#pragma unroll
    for (int j = 0; j < 4; ++j) su += sK[wv][t * 32 + m * 4 + j] * syv[wv][t * 4 + j];
    o_state[(size_t)(t0 + t) * 8 + m] = su;
  }

  // ---- A_big = diag(I - K H) into P (8 entries per lane) ----
#pragma unroll
  for (int e = 0; e < 8; ++e) {
    const int g = lane * 8 + e, m = g >> 4, n = g & 15;
    float av = 0.f;
    if ((m >> 3) == (n >> 3)) {
      const int t = m >> 3, i2 = m & 7, j2 = n & 7;
      av = (i2 == j2) ? 1.f : 0.f;
#pragma unroll
      for (int j = 0; j < 4; ++j)
        av -= sK[wv][t * 32 + i2 * 4 + j] * sH[(j + 8 * t) * 16 + (j2 + 8 * t)];
    }
    P[g] = av;
  }
  __syncthreads();

  // ---- V = A_big * Sigma ----
  acc = mm16<false>(P, C, lane);
  store_d(T, acc, lane);
  __syncthreads();

  // ---- Cov = A_big * V^T + K R K^T  (Sigma symmetric => A(ASigma)^T = A Sigma A^T) ----
  acc = mm16<true>(P, T, lane);
  if ((ln >> 3) == hi) {                           // in-block lanes only
    const int t = hi, j2 = ln & 7;
#pragma unroll
    for (int i = 0; i < 8; ++i) {
      float add = 0.f;
#pragma unroll
      for (int j = 0; j < 4; ++j)
        add += srr[j] * sK[wv][t * 32 + i * 4 + j] * sK[wv][t * 32 + j2 * 4 + j];
      acc[i] += add;
    }
  }
  __syncthreads();
  store_d(T, acc, lane);
  __syncthreads();

  // ---- coalesced cov writeout: one float4 per lane over contiguous 128 floats ----
  {
    const int g = lane * 4, t = g >> 6, m = (g >> 3) & 7, n0 = g & 7;
    const float* src = &T[(m + 8 * t) * 16 + 8 * t + n0];
    float4 v;
    v.x = src[0]; v.y = src[1]; v.z = src[2]; v.w = src[3];
    reinterpret_cast<float4*>(o_cov + (size_t)t0 * 64)[lane] = v;
  }
}

extern "C" void kernel_launch(void* const* d_in, const int* in_sizes, int n_in,
                              void* d_out, int out_size, void* d_ws, size_t ws_size,
                              hipStream_t stream) {
  (void)n_in; (void)out_size; (void)d_ws; (void)ws_size;
  const float* g_state = (const float*)d_in[0];
  const float* g_cov   = (const float*)d_in[1];
  const float* g_meas  = (const float*)d_in[2];
  const float* g_F     = (const float*)d_in[3];
  const float* g_H     = (const float*)d_in[4];
  const float* g_lq    = (const float*)d_in[5];
  const float* g_lr    = (const float*)d_in[6];

  const int B = in_sizes[0] / 8;                   // state is [B,8,1]
  float* o_state = (float*)d_out;                  // outputs concatenated flat
  float* o_cov   = o_state + (size_t)B * 8;

  const int blocks = (B + 15) / 16;                // 16 tracks per 256-thread block
  kalman_wmma_kernel<<<blocks, 256, 0, stream>>>(g_state, g_cov, g_meas, g_F, g_H,
                                                 g_lq, g_lr, o_state, o_cov, B);
}